// FlexOlmoMoE_4054449127759
// MI455X (gfx1250) — compile-verified
//
#include <hip/hip_runtime.h>
#include <math.h>

typedef __attribute__((ext_vector_type(2))) float v2f;
typedef __attribute__((ext_vector_type(8))) float v8f;

#define T_TOK 2048
#define DDIM  2048
#define FDIM  4096
#define NEXP  8

// ---------------------------------------------------------------------------
// Router: logits = x @ gate_w, softmax, top-2 -> dense combine [T, E]
// One 128-thread block per token.
// ---------------------------------------------------------------------------
__global__ __launch_bounds__(128) void router_kernel(
    const float* __restrict__ x, const float* __restrict__ gate_w,
    float* __restrict__ combine) {
  const int t   = blockIdx.x;
  const int tid = threadIdx.x;

  float acc[NEXP];
#pragma unroll
  for (int e = 0; e < NEXP; ++e) acc[e] = 0.f;

  const float* xrow = x + (size_t)t * DDIM;
  for (int d = tid; d < DDIM; d += 128) {
    const float xv = xrow[d];
    const float* g = gate_w + (size_t)d * NEXP;
#pragma unroll
    for (int e = 0; e < NEXP; ++e) acc[e] += xv * g[e];
  }

  __shared__ float red[128][NEXP];
#pragma unroll
  for (int e = 0; e < NEXP; ++e) red[tid][e] = acc[e];
  __syncthreads();
  for (int off = 64; off > 0; off >>= 1) {
    if (tid < off) {
#pragma unroll
      for (int e = 0; e < NEXP; ++e) red[tid][e] += red[tid + off][e];
    }
    __syncthreads();
  }

  if (tid == 0) {
    float lg[NEXP];
#pragma unroll
    for (int e = 0; e < NEXP; ++e) lg[e] = red[0][e];
    float mx = lg[0];
#pragma unroll
    for (int e = 1; e < NEXP; ++e) mx = fmaxf(mx, lg[e]);
    float p[NEXP], s = 0.f;
#pragma unroll
    for (int e = 0; e < NEXP; ++e) { p[e] = __expf(lg[e] - mx); s += p[e]; }
    const float inv = 1.f / s;
    // top-2 by probability, first occurrence wins ties (matches lax.top_k)
    int i1 = 0;
#pragma unroll
    for (int e = 1; e < NEXP; ++e) if (p[e] > p[i1]) i1 = e;
    int i2 = (i1 == 0) ? 1 : 0;
#pragma unroll
    for (int e = 0; e < NEXP; ++e)
      if (e != i1 && p[e] > p[i2]) i2 = e;
    float* crow = combine + (size_t)t * NEXP;
#pragma unroll
    for (int e = 0; e < NEXP; ++e)
      crow[e] = (e == i1 || e == i2) ? p[e] * inv : 0.f;
  }
}

// ---------------------------------------------------------------------------
// Phase A per expert: act[t,f] = combine[t,e] * silu(x@w1) * (x@w3)
// One wave computes a 16(M) x 32(N) tile; K-loop over D via v_wmma_f32_16x16x4_f32.
// x tile staged in LDS (coalesced load, stride-33 padding -> conflict-free).
// ---------------------------------------------------------------------------
__global__ __launch_bounds__(32) void moe_gemm1(
    const float* __restrict__ x, const float* __restrict__ w1,
    const float* __restrict__ w3, const float* __restrict__ combine,
    float* __restrict__ act, int expert) {
  const int lane = threadIdx.x;
  const int f0   = blockIdx.x * 32;
  const int t0   = blockIdx.y * 16;
  const int m16  = lane & 15;
  const int hi   = lane >> 4;  // 0: K0/K1 half, 1: K2/K3 half

  __shared__ float xs[16][33];

  v8f g0 = {}, g1 = {}, u0 = {}, u1 = {};

  for (int k0 = 0; k0 < DDIM; k0 += 32) {
#pragma unroll
    for (int i = 0; i < 16; ++i)
      xs[i][lane] = x[(size_t)(t0 + i) * DDIM + k0 + lane];
    __syncthreads();
#pragma unroll
    for (int kk = 0; kk < 8; ++kk) {
      const int kb = kk * 4;
      v2f a;
      a.x = xs[m16][kb + hi * 2 + 0];
      a.y = xs[m16][kb + hi * 2 + 1];
      const size_t rb = (size_t)(k0 + kb + hi * 2) * FDIM + f0 + m16;
      const float* w1p = w1 + rb;
      const float* w3p = w3 + rb;
      v2f b;
      b.x = w1p[0];  b.y = w1p[FDIM];
      g0 = __builtin_amdgcn_wmma_f32_16x16x4_f32(false, a, false, b, (short)0, g0, false, false);
      b.x = w1p[16]; b.y = w1p[FDIM + 16];
      g1 = __builtin_amdgcn_wmma_f32_16x16x4_f32(false, a, false, b, (short)0, g1, false, false);
      b.x = w3p[0];  b.y = w3p[FDIM];
      u0 = __builtin_amdgcn_wmma_f32_16x16x4_f32(false, a, false, b, (short)0, u0, false, false);
      b.x = w3p[16]; b.y = w3p[FDIM + 16];
      u1 = __builtin_amdgcn_wmma_f32_16x16x4_f32(false, a, false, b, (short)0, u1, false, false);
    }
    __syncthreads();
  }

  // Epilogue: SwiGLU + fold in routing weight so phase B is a plain GEMM.
#pragma unroll
  for (int r = 0; r < 8; ++r) {
    const int m = r + hi * 8;
    const int n = m16;
    const float cw = combine[(size_t)(t0 + m) * NEXP + expert];
    float g = g0[r], u = u0[r];
    float sv = g / (1.f + __expf(-g));
    act[(size_t)(t0 + m) * FDIM + f0 + n] = cw * sv * u;
    g = g1[r]; u = u1[r];
    sv = g / (1.f + __expf(-g));
    act[(size_t)(t0 + m) * FDIM + f0 + 16 + n] = cw * sv * u;
  }
}

// ---------------------------------------------------------------------------
// Phase B per expert: y (+)= act @ w2   (act already routing-scaled)
// ---------------------------------------------------------------------------
__global__ __launch_bounds__(32) void moe_gemm2(
    const float* __restrict__ act, const float* __restrict__ w2,
    float* __restrict__ y, int first) {
  const int lane = threadIdx.x;
  const int d0   = blockIdx.x * 32;
  const int t0   = blockIdx.y * 16;
  const int m16  = lane & 15;
  const int hi   = lane >> 4;

  __shared__ float hs[16][33];

  v8f c0 = {}, c1 = {};

  for (int k0 = 0; k0 < FDIM; k0 += 32) {
#pragma unroll
    for (int i = 0; i < 16; ++i)
      hs[i][lane] = act[(size_t)(t0 + i) * FDIM + k0 + lane];
    __syncthreads();
#pragma unroll
    for (int kk = 0; kk < 8; ++kk) {
      const int kb = kk * 4;
      v2f a;
      a.x = hs[m16][kb + hi * 2 + 0];
      a.y = hs[m16][kb + hi * 2 + 1];
      const float* w2p = w2 + (size_t)(k0 + kb + hi * 2) * DDIM + d0 + m16;
      v2f b;
      b.x = w2p[0];  b.y = w2p[DDIM];
      c0 = __builtin_amdgcn_wmma_f32_16x16x4_f32(false, a, false, b, (short)0, c0, false, false);
      b.x = w2p[16]; b.y = w2p[DDIM + 16];
      c1 = __builtin_amdgcn_wmma_f32_16x16x4_f32(false, a, false, b, (short)0, c1, false, false);
    }
    __syncthreads();
  }

#pragma unroll
  for (int r = 0; r < 8; ++r) {
    const int m = r + hi * 8;
    const int n = m16;
    const size_t i0 = (size_t)(t0 + m) * DDIM + d0 + n;
    if (first) {
      y[i0]      = c0[r];
      y[i0 + 16] = c1[r];
    } else {
      y[i0]      += c0[r];
      y[i0 + 16] += c1[r];
    }
  }
}

// ---------------------------------------------------------------------------
extern "C" void kernel_launch(void* const* d_in, const int* in_sizes, int n_in,
                              void* d_out, int out_size, void* d_ws, size_t ws_size,
                              hipStream_t stream) {
  (void)in_sizes; (void)n_in; (void)out_size; (void)ws_size;
  const float* x      = (const float*)d_in[0];  // [T, D]
  const float* gate_w = (const float*)d_in[1];  // [D, E]
  const float* w1     = (const float*)d_in[2];  // [E, D, F]
  const float* w3     = (const float*)d_in[3];  // [E, D, F]
  const float* w2     = (const float*)d_in[4];  // [E, F, D]
  float* y = (float*)d_out;                     // [T, D]

  float* combine = (float*)d_ws;                     // T*E floats
  float* actbuf  = combine + (size_t)T_TOK * NEXP;   // T*F floats (reused per expert)

  router_kernel<<<T_TOK, 128, 0, stream>>>(x, gate_w, combine);

  for (int e = 0; e < NEXP; ++e) {
    const float* w1e = w1 + (size_t)e * DDIM * FDIM;
    const float* w3e = w3 + (size_t)e * DDIM * FDIM;
    const float* w2e = w2 + (size_t)e * FDIM * DDIM;
    moe_gemm1<<<dim3(FDIM / 32, T_TOK / 16), 32, 0, stream>>>(
        x, w1e, w3e, combine, actbuf, e);
    moe_gemm2<<<dim3(DDIM / 32, T_TOK / 16), 32, 0, stream>>>(
        actbuf, w2e, y, e == 0 ? 1 : 0);
  }
}